// GeometricEdgeConv_75024488727153
// MI455X (gfx1250) — compile-verified
//
#include <hip/hip_runtime.h>

// GeometricEdgeConv, fused single kernel for gfx1250 (MI455X).
// Key identity: mean_k(h_ik @ W_edge) == mean_k(h_ik) @ W_edge, so the
// K=16 edge matmul collapses to one [B*N,132]x[132,128] GEMM.
// Kernel is L2/HBM bound -> fp32 WMMA (v_wmma_f32_16x16x4_f32) keeps
// reference precision at zero cost.

namespace {
constexpr int kB   = 8;
constexpr int kN   = 16384;
constexpr int kC   = 128;
constexpr int kOut = 128;
constexpr int kK   = 16;
constexpr int kEdgeIn = kC + 4;        // 132
constexpr int kRowsPerBlock = 16;      // M-tile
constexpr int kLdsStride = 132;        // %64==4 -> bank-conflict-free pair loads, 16B aligned
}

typedef float v2f __attribute__((ext_vector_type(2)));
typedef float v8f __attribute__((ext_vector_type(8)));

__global__ __launch_bounds__(256, 2)
void geoedge_fused(const float* __restrict__ x,
                   const float* __restrict__ pos,
                   const int*   __restrict__ idx,
                   const float* __restrict__ w_self,
                   const float* __restrict__ w_edge,
                   float* __restrict__ out)
{
    __shared__ float xs[kRowsPerBlock * kLdsStride];  // staged x rows   (A for w_self)
    __shared__ float mh[kRowsPerBlock * kLdsStride];  // mean edge feats (A for w_edge)
    __shared__ float red[256 * 4];                    // rel_pos/dist reduction scratch

    const int tid  = threadIdx.x;                 // 0..255
    const int row0 = blockIdx.x * kRowsPerBlock;  // global row of tile
    const int b    = row0 / kN;                   // tile never straddles batches
    const int n0   = row0 - b * kN;

    //=== Phase A1: stage own rows + gather/mean neighbor features =========
    {
        const int r  = tid >> 4;          // row in tile, 16 threads per row
        const int cg = (tid & 15) << 3;   // 8-column slice
        const int rowbase = (b * kN + n0 + r) * kC;

        const float4 s0 = *(const float4*)(x + rowbase + cg);
        const float4 s1 = *(const float4*)(x + rowbase + cg + 4);
        *(float4*)(xs + r * kLdsStride + cg)     = s0;
        *(float4*)(xs + r * kLdsStride + cg + 4) = s1;

        float a0=0.f,a1=0.f,a2=0.f,a3=0.f,a4=0.f,a5=0.f,a6=0.f,a7=0.f;
        const int ibase = (b * kN + n0 + r) * kK;
        int j = idx[ibase];
        #pragma unroll
        for (int k = 0; k < kK; ++k) {
            const float* nb = x + (b * kN + j) * kC + cg;
            const int jn = (k + 1 < kK) ? idx[ibase + k + 1] : j;
            __builtin_prefetch(x + (b * kN + jn) * kC + cg, 0, 0);
            const float4 p0 = *(const float4*)(nb);
            const float4 p1 = *(const float4*)(nb + 4);
            a0 += p0.x; a1 += p0.y; a2 += p0.z; a3 += p0.w;
            a4 += p1.x; a5 += p1.y; a6 += p1.z; a7 += p1.w;
            j = jn;
        }
        const float inv = 1.0f / (float)kK;
        float4 m0; m0.x = a0*inv; m0.y = a1*inv; m0.z = a2*inv; m0.w = a3*inv;
        float4 m1; m1.x = a4*inv; m1.y = a5*inv; m1.z = a6*inv; m1.w = a7*inv;
        *(float4*)(mh + r * kLdsStride + cg)     = m0;
        *(float4*)(mh + r * kLdsStride + cg + 4) = m1;
    }

    //=== Phase A2: mean rel_pos (3) and mean dist^2 (1) ====================
    {
        const int r = tid >> 4;
        const int k = tid & 15;
        const int pbase = (b * kN + n0 + r) * 3;
        const int j     = idx[(b * kN + n0 + r) * kK + k];
        const int qbase = (b * kN + j) * 3;
        const float rx = pos[pbase + 0] - pos[qbase + 0];
        const float ry = pos[pbase + 1] - pos[qbase + 1];
        const float rz = pos[pbase + 2] - pos[qbase + 2];
        const float d  = rx*rx + ry*ry + rz*rz;
        float* rr = red + tid * 4;
        rr[0] = rx; rr[1] = ry; rr[2] = rz; rr[3] = d;
    }
    __syncthreads();
    if (tid < kRowsPerBlock) {
        float sx=0.f, sy=0.f, sz=0.f, sd=0.f;
        #pragma unroll
        for (int k = 0; k < kK; ++k) {
            const float* rr = red + (tid * kK + k) * 4;
            sx += rr[0]; sy += rr[1]; sz += rr[2]; sd += rr[3];
        }
        const float inv = 1.0f / (float)kK;
        float* mrow = mh + tid * kLdsStride + kC;
        mrow[0] = sx*inv; mrow[1] = sy*inv; mrow[2] = sz*inv; mrow[3] = sd*inv;
    }
    __syncthreads();

    //=== Phase B: per-wave WMMA, 16(M) x 16(N) tile per wave ===============
    const int wave = tid >> 5;        // 0..7 -> OUT column tile
    const int lane = tid & 31;
    const int half = lane >> 4;       // 0 | 1
    const int l15  = lane & 15;
    const int col0 = wave << 4;
    const int koff = 2 * half;        // A/B fragment K sublane offset

    v8f acc = {0.f,0.f,0.f,0.f,0.f,0.f,0.f,0.f};

    const float* xrow = xs + l15 * kLdsStride;
    const float* mrow = mh + l15 * kLdsStride;

    #pragma unroll 8
    for (int kk = 0; kk < kC; kk += 4) {
        const v2f a = *(const v2f*)(xrow + kk + koff);
        v2f bw;
        bw.x = w_self[(kk + koff + 0) * kOut + col0 + l15];
        bw.y = w_self[(kk + koff + 1) * kOut + col0 + l15];
        acc = __builtin_amdgcn_wmma_f32_16x16x4_f32(
                  false, a, false, bw, (short)0, acc, false, false);
    }
    #pragma unroll 8
    for (int kk = 0; kk < kEdgeIn; kk += 4) {
        const v2f a = *(const v2f*)(mrow + kk + koff);
        v2f bw;
        bw.x = w_edge[(kk + koff + 0) * kOut + col0 + l15];
        bw.y = w_edge[(kk + koff + 1) * kOut + col0 + l15];
        acc = __builtin_amdgcn_wmma_f32_16x16x4_f32(
                  false, a, false, bw, (short)0, acc, false, false);
    }

    //=== Epilogue: leaky_relu(0.2) + coalesced store =======================
    float* orow = out + (size_t)(row0 + 8 * half) * kOut + col0 + l15;
    #pragma unroll
    for (int jj = 0; jj < 8; ++jj) {
        float v = acc[jj];
        v = fmaxf(v, 0.2f * v);           // leaky relu, slope 0.2
        orow[(size_t)jj * kOut] = v;
    }
}

extern "C" void kernel_launch(void* const* d_in, const int* in_sizes, int n_in,
                              void* d_out, int out_size, void* d_ws, size_t ws_size,
                              hipStream_t stream) {
    (void)in_sizes; (void)n_in; (void)out_size; (void)d_ws; (void)ws_size;
    const float* x      = (const float*)d_in[0];
    const float* pos    = (const float*)d_in[1];
    const int*   idx    = (const int*)  d_in[2];
    const float* w_self = (const float*)d_in[3];
    const float* w_edge = (const float*)d_in[4];
    float* out = (float*)d_out;

    const int mTiles = (kB * kN) / kRowsPerBlock;   // 8192
    geoedge_fused<<<dim3(mTiles), dim3(256), 0, stream>>>(
        x, pos, idx, w_self, w_edge, out);
}